// SelfCrossAttn_53687091200227
// MI455X (gfx1250) — compile-verified
//
#include <hip/hip_runtime.h>
#include <math.h>

// MI455X / gfx1250, wave32. fp32 WMMA (V_WMMA_F32_16X16X4_F32) GEMM pipeline
// with double-buffered LDS and async global->LDS staging (ASYNCcnt).

typedef __attribute__((ext_vector_type(2))) float v2f;
typedef __attribute__((ext_vector_type(8))) float v8f;

#define AS_LD 68   // 64 + 4 pad (words) to avoid LDS bank conflicts
#define BS_LD 132  // 128 + 4 pad

#if __has_builtin(__builtin_amdgcn_global_load_async_to_lds_b128)
#define HAS_ASYNC_LDS 1
#else
#define HAS_ASYNC_LDS 0
#endif

// Builtin signature (from hipcc diagnostic): param0 = <4 x i32> addrspace(1)*,
// param1 = <4 x i32> addrspace(3)*, then imm offset, imm cpol.
typedef int v4i __attribute__((vector_size(16)));
typedef __attribute__((address_space(1))) v4i ga_v4i;
typedef __attribute__((address_space(3))) v4i ls_v4i;

// 16-byte memory->LDS copy: async on gfx1250 (GLOBAL_LOAD_ASYNC_TO_LDS_B128),
// synchronous fallback otherwise.
__device__ __forceinline__ void async_copy_b128(const float* gsrc, float* ldst) {
#if HAS_ASYNC_LDS
  __builtin_amdgcn_global_load_async_to_lds_b128((ga_v4i*)gsrc, (ls_v4i*)ldst,
                                                 0, 0);
#else
  *(float4*)ldst = *(const float4*)gsrc;
#endif
}

__device__ __forceinline__ void async_wait_all() {
#if HAS_ASYNC_LDS
#if __has_builtin(__builtin_amdgcn_s_wait_asynccnt)
  __builtin_amdgcn_s_wait_asynccnt(0);
#else
  asm volatile("s_wait_asynccnt 0" ::: "memory");
#endif
#endif
}

// C(M x N) = A(M x K) * B(K x N), fp32, block tile 64(M) x 128(N), K-step 16.
// TRANS_A: A stored K x M row-major (need A[k][m] = Asrc[k*lda + m])
// TRANS_B: B stored N x K row-major (need B[k][n] = Bsrc[n*ldb + k])
// EPI: 0 = plain store, 1 = val *= alpha, 2 = val += bias[row] + resid[...]
template <int TRANS_A, int TRANS_B, int EPI>
__global__ __launch_bounds__(256) void gemm_wmma_f32(
    const float* __restrict__ A, long long Abs,
    const float* __restrict__ B, long long Bbs,
    float* __restrict__ C, long long Cbs,
    int K, int lda, int ldb, int ldc,
    float alpha,
    const float* __restrict__ bias,
    const float* __restrict__ resid, long long Rbs) {
  __shared__ __align__(16) float As[2][16 * AS_LD];
  __shared__ __align__(16) float Bs[2][16 * BS_LD];

  A += (size_t)blockIdx.z * Abs;
  B += (size_t)blockIdx.z * Bbs;
  C += (size_t)blockIdx.z * Cbs;

  const int tid   = threadIdx.x;  // 0..255, 8 waves
  const int lane  = tid & 31;
  const int wave  = tid >> 5;
  const int half  = lane >> 4;    // lanes 0-15 -> K pair {0,1}, 16-31 -> {2,3}
  const int ln    = lane & 15;
  const int waveM = wave >> 2;    // 0..1
  const int waveN = wave & 3;     // 0..3
  const int blockM = blockIdx.y * 64;
  const int blockN = blockIdx.x * 128;

  v8f acc[2][2] = {};

  // Stage one 16-wide K slab of A and B into LDS buffer `buf`.
  auto stage = [&](int buf, int kb) {
    if (TRANS_A == 0) {
      // transpose scatter (row-major M x K -> As[k][m]); synchronous
      const int m  = tid >> 2;           // 0..63
      const int k4 = (tid & 3) << 2;     // 0,4,8,12
      float4 av = *(const float4*)(A + (size_t)(blockM + m) * lda + kb + k4);
      As[buf][(k4 + 0) * AS_LD + m] = av.x;
      As[buf][(k4 + 1) * AS_LD + m] = av.y;
      As[buf][(k4 + 2) * AS_LD + m] = av.z;
      As[buf][(k4 + 3) * AS_LD + m] = av.w;
    } else {
      // A already K-major: contiguous row copy -> async to LDS
      const int kk = tid >> 4;           // 0..15
      const int m4 = (tid & 15) << 2;    // 0..60
      async_copy_b128(A + (size_t)(kb + kk) * lda + blockM + m4,
                      &As[buf][kk * AS_LD + m4]);
    }
    if (TRANS_B == 0) {
      // B K-major: contiguous row copy -> async to LDS
#pragma unroll
      for (int r = 0; r < 2; ++r) {
        const int idx = tid + r * 256;
        const int kk  = idx >> 5;        // 0..15
        const int n4  = (idx & 31) << 2; // 0..124
        async_copy_b128(B + (size_t)(kb + kk) * ldb + blockN + n4,
                        &Bs[buf][kk * BS_LD + n4]);
      }
    } else {
      // transpose scatter (row-major N x K -> Bs[k][n]); synchronous
      const int n  = tid >> 1;           // 0..127
      const int k8 = (tid & 1) << 3;     // 0,8
      const float* src = B + (size_t)(blockN + n) * ldb + kb + k8;
      float4 b0 = *(const float4*)(src);
      float4 b1 = *(const float4*)(src + 4);
      Bs[buf][(k8 + 0) * BS_LD + n] = b0.x;
      Bs[buf][(k8 + 1) * BS_LD + n] = b0.y;
      Bs[buf][(k8 + 2) * BS_LD + n] = b0.z;
      Bs[buf][(k8 + 3) * BS_LD + n] = b0.w;
      Bs[buf][(k8 + 4) * BS_LD + n] = b1.x;
      Bs[buf][(k8 + 5) * BS_LD + n] = b1.y;
      Bs[buf][(k8 + 6) * BS_LD + n] = b1.z;
      Bs[buf][(k8 + 7) * BS_LD + n] = b1.w;
    }
  };

  stage(0, 0);
  const int NKB = K >> 4;
  int cur = 0;
  for (int kbi = 0; kbi < NKB; ++kbi, cur ^= 1) {
    if (TRANS_A == 1 || TRANS_B == 0) async_wait_all();
    __syncthreads();  // buffer `cur` fully staged; everyone done with `cur^1`
    if (kbi + 1 < NKB) stage(cur ^ 1, (kbi + 1) << 4);  // prefetch next slab

    // 4 WMMA K-steps of 4 over this 16-wide K slab
#pragma unroll
    for (int ks = 0; ks < 4; ++ks) {
      v2f a[2], bb[2];
#pragma unroll
      for (int mi = 0; mi < 2; ++mi) {
        const int m = waveM * 32 + mi * 16 + ln;
        a[mi].x = As[cur][(ks * 4 + half * 2 + 0) * AS_LD + m];
        a[mi].y = As[cur][(ks * 4 + half * 2 + 1) * AS_LD + m];
      }
#pragma unroll
      for (int ni = 0; ni < 2; ++ni) {
        const int n = waveN * 32 + ni * 16 + ln;
        bb[ni].x = Bs[cur][(ks * 4 + half * 2 + 0) * BS_LD + n];
        bb[ni].y = Bs[cur][(ks * 4 + half * 2 + 1) * BS_LD + n];
      }
#pragma unroll
      for (int mi = 0; mi < 2; ++mi)
#pragma unroll
        for (int ni = 0; ni < 2; ++ni)
          acc[mi][ni] = __builtin_amdgcn_wmma_f32_16x16x4_f32(
              false, a[mi], false, bb[ni], (short)0, acc[mi][ni], false, false);
    }
  }

  // C/D fragment layout: VGPR r -> row (half*8 + r), lane ln -> col
#pragma unroll
  for (int mi = 0; mi < 2; ++mi) {
#pragma unroll
    for (int ni = 0; ni < 2; ++ni) {
#pragma unroll
      for (int r = 0; r < 8; ++r) {
        const int row = blockM + waveM * 32 + mi * 16 + half * 8 + r;
        const int col = blockN + waveN * 32 + ni * 16 + ln;
        float val = acc[mi][ni][r];
        if (EPI == 1) val *= alpha;
        if (EPI == 2)
          val += bias[row] +
                 resid[(size_t)blockIdx.z * Rbs + (size_t)row * ldc + col];
        C[(size_t)row * ldc + col] = val;
      }
    }
  }
}

// Row softmax over 512-wide rows; one wave32 per row, 8 rows per block.
__global__ __launch_bounds__(256) void softmax_rows_512(float* __restrict__ S) {
  const int row  = blockIdx.x * 8 + (threadIdx.x >> 5);
  const int lane = threadIdx.x & 31;
  float* p = S + (size_t)row * 512;
  float vals[16];
  float m = -3.4e38f;
#pragma unroll
  for (int j = 0; j < 16; ++j) {
    vals[j] = p[lane + 32 * j];
    m = fmaxf(m, vals[j]);
  }
#pragma unroll
  for (int off = 16; off > 0; off >>= 1) m = fmaxf(m, __shfl_xor(m, off, 32));
  float s = 0.f;
#pragma unroll
  for (int j = 0; j < 16; ++j) {
    vals[j] = __expf(vals[j] - m);
    s += vals[j];
  }
#pragma unroll
  for (int off = 16; off > 0; off >>= 1) s += __shfl_xor(s, off, 32);
  const float inv = 1.0f / s;
#pragma unroll
  for (int j = 0; j < 16; ++j) p[lane + 32 * j] = vals[j] * inv;
}

extern "C" void kernel_launch(void* const* d_in, const int* in_sizes, int n_in,
                              void* d_out, int out_size, void* d_ws,
                              size_t ws_size, hipStream_t stream) {
  const float* x     = (const float*)d_in[0];
  const float* Wq    = (const float*)d_in[1];
  const float* Wk    = (const float*)d_in[2];
  const float* Wv    = (const float*)d_in[3];
  const float* Wproj = (const float*)d_in[4];
  const float* bproj = (const float*)d_in[5];
  float* out = (float*)d_out;

  const int b = 4, c = 512, Nn = 4096, H = 4, E = Nn / H;  // E = 1024
  const long long bcN = (long long)c * Nn;                 // per-batch elems
  const long long chE = (long long)c * E;                  // per-(b,h) block

  float* q = (float*)d_ws;          // (b,c,N) == (b,H,c,E) contiguous
  float* k = q + (size_t)b * bcN;
  float* v = k + (size_t)b * bcN;
  float* S = v + (size_t)b * bcN;   // (b*H, 512, 512) attention logits/probs
  float* U = q;                     // attn output (per (b,h): E x c) aliases q

  const float scale = 0.08838834764831845f;  // (c/H)^-0.5 = 128^-0.5
  dim3 blk(256);

  // 1) Q/K/V: per batch, W(512x512) x X(512x4096)
  dim3 g1(Nn / 128, c / 64, b);
  gemm_wmma_f32<0, 0, 0><<<g1, blk, 0, stream>>>(Wq, 0, x, bcN, q, bcN,
                                                 c, c, Nn, Nn, 1.f, nullptr, nullptr, 0);
  gemm_wmma_f32<0, 0, 0><<<g1, blk, 0, stream>>>(Wk, 0, x, bcN, k, bcN,
                                                 c, c, Nn, Nn, 1.f, nullptr, nullptr, 0);
  gemm_wmma_f32<0, 0, 0><<<g1, blk, 0, stream>>>(Wv, 0, x, bcN, v, bcN,
                                                 c, c, Nn, Nn, 1.f, nullptr, nullptr, 0);

  // 2) S = scale * Qh(512xE) * Kh(512xE)^T per (b,h)
  dim3 g2(512 / 128, 512 / 64, b * H);
  gemm_wmma_f32<0, 1, 1><<<g2, blk, 0, stream>>>(q, chE, k, chE, S, 512LL * 512,
                                                 E, E, E, 512, scale, nullptr, nullptr, 0);

  // 3) row softmax over 512-wide rows (b*H*512 = 8192 rows)
  softmax_rows_512<<<dim3((b * H * 512) / 8), blk, 0, stream>>>(S);

  // 4) Ot(E x c) = Vh^T * P^T : TRANS_A (Vh is c x E) + TRANS_B (P is c x c);
  //    row-major (e, cc) store per (b,h) == swapaxes(2,3).reshape result.
  dim3 g3(512 / 128, E / 64, b * H);
  gemm_wmma_f32<1, 1, 0><<<g3, blk, 0, stream>>>(v, chE, S, 512LL * 512, U, chE,
                                                 c, E, 512, c, 1.f, nullptr, nullptr, 0);

  // 5) out = Wproj * U + bproj + x  (per batch)
  dim3 g4(Nn / 128, c / 64, b);
  gemm_wmma_f32<0, 0, 2><<<g4, blk, 0, stream>>>(Wproj, 0, U, bcN, out, bcN,
                                                 c, c, Nn, Nn, 1.f, bproj, x, bcN);
}